// TeacherGCN_66073776882319
// MI455X (gfx1250) — compile-verified
//
#include <hip/hip_runtime.h>

// ---------------------------------------------------------------------------
// 3-layer GCN for MI455X (gfx1250, wave32).
//  - GEMMs via V_WMMA_F32_16X16X4_F32 (fp32-exact, matches reference dtype)
//  - Edge aggregation via wave-per-edge global_atomic_add_f32 (L2-resident)
// ---------------------------------------------------------------------------

typedef __attribute__((ext_vector_type(2))) float v2f;
typedef __attribute__((ext_vector_type(8))) float v8f;

#define DIN 128
#define DH  128
#define DOUT 64

__device__ __forceinline__ void atomAddF(float* p, float v) {
    unsafeAtomicAdd(p, v);   // global_atomic_add_f32 (no CAS loop)
}

// ------------------------- degree / normalization --------------------------

__global__ void init_deg_kernel(float* __restrict__ deg, int n) {
    int i = blockIdx.x * blockDim.x + threadIdx.x;
    if (i < n) deg[i] = 1.0f;                 // self loop
}

__global__ void deg_kernel(const int* __restrict__ dst, float* __restrict__ deg, int e) {
    int i = blockIdx.x * blockDim.x + threadIdx.x;
    if (i < e) atomAddF(&deg[dst[i]], 1.0f);
}

__global__ void dinv_kernel(float* __restrict__ deg, int n) {
    int i = blockIdx.x * blockDim.x + threadIdx.x;
    if (i < n) deg[i] = __frsqrt_rn(deg[i]);  // deg >= 1 always
}

__global__ void zero_kernel(float* __restrict__ p, int n) {
    int i = blockIdx.x * blockDim.x + threadIdx.x;
    int stride = gridDim.x * blockDim.x;
    for (; i < n; i += stride) p[i] = 0.0f;
}

// ------------------------------- WMMA GEMM ---------------------------------
// Y[M x N] = X[M x K] * W[K x N], row-major, one wave per 16x16 tile.
// A frag (16x4 f32): lane&15 = row M, lanes>=16 hold K+2/K+3.
// B frag (4x16 f32): lane&15 = col N, lanes>=16 hold K+2/K+3.
// C/D (16x16 f32):  VGPR r -> row r + 8*(lane>=16), col lane&15.
__global__ __launch_bounds__(256)
void gemm_wmma_f32(const float* __restrict__ X, const float* __restrict__ W,
                   float* __restrict__ Y, int M, int K, int N) {
    int wave  = (blockIdx.x * blockDim.x + threadIdx.x) >> 5;
    int lane  = threadIdx.x & 31;
    int tilesN = N >> 4;
    int tm = wave / tilesN;
    int tn = wave - tm * tilesN;
    if (tm * 16 >= M) return;                 // wave-uniform: EXEC stays all-1s

    int half = lane >> 4;                     // 0: K+{0,1}, 1: K+{2,3}
    int lr   = lane & 15;                     // A row / B col within tile
    int koff = half << 1;
    int m0 = tm << 4, n0 = tn << 4;

    v8f c = {};
    const float* xrow = X + (m0 + lr) * K + koff;
    const float* wcol = W + koff * N + n0 + lr;
    #pragma unroll 8
    for (int k = 0; k < K; k += 4) {
        v2f a, b;
        a.x = xrow[k];         a.y = xrow[k + 1];
        b.x = wcol[k * N];     b.y = wcol[(k + 1) * N];
        c = __builtin_amdgcn_wmma_f32_16x16x4_f32(
                false, a, false, b, (short)0, c, false, false);
    }
    #pragma unroll
    for (int r = 0; r < 8; ++r)
        Y[(m0 + r + half * 8) * N + n0 + lr] = c[r];
}

// ---------------------------- edge aggregation -----------------------------
// One wave per edge; 32 lanes x float4 = 128 floats (or float2 = 64 floats).

__global__ __launch_bounds__(256)
void scatter_d128(const float* __restrict__ h, const int* __restrict__ src,
                  const int* __restrict__ dst, const float* __restrict__ dinv,
                  float* __restrict__ out, int e) {
    int w = (blockIdx.x * blockDim.x + threadIdx.x) >> 5;
    int lane = threadIdx.x & 31;
    if (w >= e) return;
    int s = src[w], d = dst[w];
    float norm = dinv[s] * dinv[d];
    const float4 v = *(const float4*)(h + (size_t)s * 128 + lane * 4);
    float* o = out + (size_t)d * 128 + lane * 4;
    atomAddF(o + 0, norm * v.x);
    atomAddF(o + 1, norm * v.y);
    atomAddF(o + 2, norm * v.z);
    atomAddF(o + 3, norm * v.w);
}

__global__ __launch_bounds__(256)
void scatter_d64(const float* __restrict__ h, const int* __restrict__ src,
                 const int* __restrict__ dst, const float* __restrict__ dinv,
                 float* __restrict__ out, int e) {
    int w = (blockIdx.x * blockDim.x + threadIdx.x) >> 5;
    int lane = threadIdx.x & 31;
    if (w >= e) return;
    int s = src[w], d = dst[w];
    float norm = dinv[s] * dinv[d];
    const float2 v = *(const float2*)(h + (size_t)s * 64 + lane * 2);
    float* o = out + (size_t)d * 64 + lane * 2;
    atomAddF(o + 0, norm * v.x);
    atomAddF(o + 1, norm * v.y);
}

// acc += dinv[v]^2 * h (self loop) + bias; optional ReLU
__global__ void finalize_kernel(float* __restrict__ acc, const float* __restrict__ h,
                                const float* __restrict__ dinv,
                                const float* __restrict__ bias,
                                int n, int dfeat, int relu) {
    int i = blockIdx.x * blockDim.x + threadIdx.x;
    if (i >= n * dfeat) return;
    int v = i / dfeat;
    int f = i - v * dfeat;
    float di = dinv[v];
    float r = acc[i] + di * di * h[i] + bias[f];
    if (relu) r = fmaxf(r, 0.0f);
    acc[i] = r;
}

// -------------------------------- launcher ---------------------------------

extern "C" void kernel_launch(void* const* d_in, const int* in_sizes, int n_in,
                              void* d_out, int out_size, void* d_ws, size_t ws_size,
                              hipStream_t stream) {
    const float* x   = (const float*)d_in[0];
    const int*   ei  = (const int*)d_in[1];
    const float* W1  = (const float*)d_in[2];
    const float* b1  = (const float*)d_in[3];
    const float* W2  = (const float*)d_in[4];
    const float* b2  = (const float*)d_in[5];
    const float* W3  = (const float*)d_in[6];
    const float* b3  = (const float*)d_in[7];
    float* out = (float*)d_out;

    const int N = in_sizes[0] / DIN;       // 50000
    const int E = in_sizes[1] / 2;         // 600000
    const int* src = ei;
    const int* dstp = ei + E;

    float* bufA = (float*)d_ws;                       // N x 128 (gemm output h)
    float* bufB = bufA + (size_t)N * DH;              // N x 128 (aggregation acc)
    float* dinv = bufB + (size_t)N * DH;              // N      (deg -> dinv)

    const int T = 256;
    const int nodeBlocks  = (N + T - 1) / T;
    const int edgeBlocks  = (E + T - 1) / T;
    const int edgeWaveBlk = (E * 32 + T - 1) / T;     // one wave per edge
    const int f128Blocks  = (N * DH + T - 1) / T;
    const int f64Blocks   = (N * DOUT + T - 1) / T;
    const int gemmBlk128  = ((N / 16) * (DH / 16)   * 32 + T - 1) / T;
    const int gemmBlk64   = ((N / 16) * (DOUT / 16) * 32 + T - 1) / T;

    // normalization: deg = 1 + indeg;  dinv = rsqrt(deg)
    init_deg_kernel<<<nodeBlocks, T, 0, stream>>>(dinv, N);
    deg_kernel<<<edgeBlocks, T, 0, stream>>>(dstp, dinv, E);
    dinv_kernel<<<nodeBlocks, T, 0, stream>>>(dinv, N);

    // ---- layer 1: h = x@W1 ; agg ; +b1, ReLU ----
    gemm_wmma_f32<<<gemmBlk128, T, 0, stream>>>(x, W1, bufA, N, DIN, DH);
    zero_kernel<<<2048, T, 0, stream>>>(bufB, N * DH);
    scatter_d128<<<edgeWaveBlk, T, 0, stream>>>(bufA, src, dstp, dinv, bufB, E);
    finalize_kernel<<<f128Blocks, T, 0, stream>>>(bufB, bufA, dinv, b1, N, DH, 1);

    // ---- layer 2: h = bufB@W2 ; agg ; +b2, ReLU ----
    gemm_wmma_f32<<<gemmBlk128, T, 0, stream>>>(bufB, W2, bufA, N, DH, DH);
    zero_kernel<<<2048, T, 0, stream>>>(bufB, N * DH);
    scatter_d128<<<edgeWaveBlk, T, 0, stream>>>(bufA, src, dstp, dinv, bufB, E);
    finalize_kernel<<<f128Blocks, T, 0, stream>>>(bufB, bufA, dinv, b2, N, DH, 1);

    // ---- layer 3: h = bufB@W3 ; agg ; +b3 (no ReLU) ----
    gemm_wmma_f32<<<gemmBlk64, T, 0, stream>>>(bufB, W3, bufA, N, DH, DOUT);
    zero_kernel<<<2048, T, 0, stream>>>(out, N * DOUT);
    scatter_d64<<<edgeWaveBlk, T, 0, stream>>>(bufA, src, dstp, dinv, out, E);
    finalize_kernel<<<f64Blocks, T, 0, stream>>>(out, bufA, dinv, b3, N, DOUT, 0);
}